// GATAttention_38732015075671
// MI455X (gfx1250) — compile-verified
//
#include <hip/hip_runtime.h>

typedef __attribute__((ext_vector_type(2))) float v2f;
typedef __attribute__((ext_vector_type(4))) float v4f;
typedef __attribute__((ext_vector_type(8))) float v8f;

#define N_NODES 1024
#define DDIM 128
#define HDIM 128

// Workspace layout (floats):
//   ws[0   ..127 ]  v_edge  = Wr.T @ w3
//   ws[128 ..255 ]  v_node  = Wh.T @ w1 + Wt.T @ w2
//   ws[256]         cst     = bh@w1 + bt@w2 + br@w3 + wt_b
//   ws[512 ..1535]  npb[j]  = node_feat[j]·v_node + cst

__global__ void gat_prep_vec(const float* __restrict__ Wh, const float* __restrict__ bh,
                             const float* __restrict__ Wt, const float* __restrict__ bt,
                             const float* __restrict__ Wr, const float* __restrict__ br,
                             const float* __restrict__ wtw, const float* __restrict__ wtb,
                             float* __restrict__ ws) {
  const int d = threadIdx.x;  // 128 threads
  float ve = 0.0f, vn = 0.0f;
  for (int h = 0; h < HDIM; ++h) {
    const float w1 = wtw[h], w2 = wtw[HDIM + h], w3 = wtw[2 * HDIM + h];
    vn = fmaf(Wh[h * DDIM + d], w1, vn);
    vn = fmaf(Wt[h * DDIM + d], w2, vn);
    ve = fmaf(Wr[h * DDIM + d], w3, ve);
  }
  ws[d] = ve;
  ws[DDIM + d] = vn;
  if (d == 0) {
    float c = wtb[0];
    for (int h = 0; h < HDIM; ++h)
      c += bh[h] * wtw[h] + bt[h] * wtw[HDIM + h] + br[h] * wtw[2 * HDIM + h];
    ws[2 * DDIM] = c;
  }
}

__global__ void gat_prep_npb(const float* __restrict__ node_feat,
                             const float* __restrict__ ws,
                             float* __restrict__ npb) {
  const int j = blockIdx.x * blockDim.x + threadIdx.x;  // 4 x 256 = 1024
  const float* __restrict__ vn = ws + DDIM;
  const float* __restrict__ nf = node_feat + j * DDIM;
  float acc = ws[2 * DDIM];
#pragma unroll 8
  for (int d = 0; d < DDIM; ++d) acc = fmaf(nf[d], vn[d], acc);
  npb[j] = acc;
}

// One block per row i. 8 waves; each wave computes 8 tiles of 16 columns via
// V_WMMA_F32_16X16X4_F32 (A = broadcast v_edge chunk, B = edge_feat tile),
// then a fused block-wide softmax over the 1024 energies.
__global__ void __launch_bounds__(256) gat_main(const float* __restrict__ edge,
                                                const float* __restrict__ ws,
                                                float* __restrict__ out) {
  __shared__ float s_v[DDIM];
  __shared__ float s_e[N_NODES];
  __shared__ float s_red[16];

  const int tid = threadIdx.x;
  const int i = blockIdx.x;
  if (tid < DDIM) s_v[tid] = ws[tid];
  __syncthreads();

  const int wave = tid >> 5;
  const int lane = tid & 31;
  const int half = lane >> 4;   // 0: supplies k={0,1}, 1: supplies k={2,3}
  const int ln   = lane & 15;   // column (j) within tile
  const float* __restrict__ erow = edge + (size_t)i * (N_NODES * DDIM);

  for (int t = wave; t < N_NODES / 16; t += 8) {
    const float* __restrict__ p = erow + (size_t)(t * 16 + ln) * DDIM + half * 2;
    v8f acc = {};
#pragma unroll
    for (int c = 0; c < DDIM / 4; ++c) {
      v2f b = *(const v2f*)(p + 4 * c);                    // edge[i, j0+ln, 4c + 2*half + {0,1}]
      v2f a = *(const v2f*)(&s_v[4 * c + half * 2]);       // v_edge same k-chunk (broadcast over M)
      acc = __builtin_amdgcn_wmma_f32_16x16x4_f32(
          /*neg_a=*/false, a, /*neg_b=*/false, b,
          /*c_mod=*/(short)0, acc, /*reuse_a=*/false, /*reuse_b=*/false);
    }
    // D[m,n] is independent of m; lane holds column n = ln in every acc element.
    if (half == 0) s_e[t * 16 + ln] = acc[0];
  }
  __syncthreads();

  // Fused softmax over the row: thread tid owns j = 4*tid .. 4*tid+3.
  const float* __restrict__ npb = ws + 512;
  v4f ev = *(const v4f*)(&s_e[tid * 4]);
  v4f np = *(const v4f*)(&npb[tid * 4]);
  ev += np;

  float m = fmaxf(fmaxf(ev.x, ev.y), fmaxf(ev.z, ev.w));
  for (int o = 16; o > 0; o >>= 1) m = fmaxf(m, __shfl_xor(m, o));
  if (lane == 0) s_red[wave] = m;
  __syncthreads();
  float M = s_red[0];
#pragma unroll
  for (int w = 1; w < 8; ++w) M = fmaxf(M, s_red[w]);

  v4f ex;
  ex.x = __expf(ev.x - M);
  ex.y = __expf(ev.y - M);
  ex.z = __expf(ev.z - M);
  ex.w = __expf(ev.w - M);
  float s = ex.x + ex.y + ex.z + ex.w;
  for (int o = 16; o > 0; o >>= 1) s += __shfl_xor(s, o);
  if (lane == 0) s_red[8 + wave] = s;
  __syncthreads();
  float S = 0.0f;
#pragma unroll
  for (int w = 0; w < 8; ++w) S += s_red[8 + w];

  ex *= (1.0f / S);
  *(v4f*)(out + (size_t)i * N_NODES + tid * 4) = ex;
}

extern "C" void kernel_launch(void* const* d_in, const int* in_sizes, int n_in,
                              void* d_out, int out_size, void* d_ws, size_t ws_size,
                              hipStream_t stream) {
  const float* node_feat = (const float*)d_in[0];
  const float* edge_feat = (const float*)d_in[1];
  // d_in[2] (mask) is unused by the reference.
  const float* Wh  = (const float*)d_in[3];
  const float* bh  = (const float*)d_in[4];
  const float* Wt  = (const float*)d_in[5];
  const float* bt  = (const float*)d_in[6];
  const float* Wr  = (const float*)d_in[7];
  const float* br  = (const float*)d_in[8];
  const float* wtw = (const float*)d_in[9];
  const float* wtb = (const float*)d_in[10];
  float* ws  = (float*)d_ws;
  float* out = (float*)d_out;

  gat_prep_vec<<<1, 128, 0, stream>>>(Wh, bh, Wt, bt, Wr, br, wtw, wtb, ws);
  gat_prep_npb<<<4, 256, 0, stream>>>(node_feat, ws, ws + 512);
  gat_main<<<N_NODES, 256, 0, stream>>>(edge_feat, ws, out);
}